// NearestTranslationEmbStack_72808285602353
// MI455X (gfx1250) — compile-verified
//
#include <hip/hip_runtime.h>

typedef __attribute__((ext_vector_type(2))) float v2f;
typedef __attribute__((ext_vector_type(8))) float v8f;

#define NPTS 8192
#define NFEAT 256

// ---------------------------------------------------------------------------
// Kernel 1: pack WMMA operand tiles.
//   Apack[n][0..3] = { -2*x1, -2*y1, -2*z1, |t1|^2 }   (A matrix rows, refs)
//   Bpack[n][0..3] = {   x2,    y2,    z2,   1.0   }   (B matrix cols, queries)
// Score S[n1][n2] = A[n1]·B[n2] = |t1|^2 - 2*t1·t2  (|t2|^2 dropped: constant
// per query, does not change the argmin over n1).
// ---------------------------------------------------------------------------
__global__ void nn_prep_kernel(const float* __restrict__ t1,
                               const float* __restrict__ t2,
                               float* __restrict__ Apack,
                               float* __restrict__ Bpack) {
    int i = blockIdx.x * blockDim.x + threadIdx.x;
    if (i >= NPTS) return;
    float x1 = t1[i], y1 = t1[NPTS + i], z1 = t1[2 * NPTS + i];
    Apack[i * 4 + 0] = -2.0f * x1;
    Apack[i * 4 + 1] = -2.0f * y1;
    Apack[i * 4 + 2] = -2.0f * z1;
    Apack[i * 4 + 3] = x1 * x1 + y1 * y1 + z1 * z1;
    float x2 = t2[i], y2 = t2[NPTS + i], z2 = t2[2 * NPTS + i];
    Bpack[i * 4 + 0] = x2;
    Bpack[i * 4 + 1] = y2;
    Bpack[i * 4 + 2] = z2;
    Bpack[i * 4 + 3] = 1.0f;
}

// ---------------------------------------------------------------------------
// Kernel 2: WMMA-based argmin. One wave (32 lanes) owns 16 query columns.
// Per iteration: load a 16x4 f32 A tile (refs), multiply against the
// loop-invariant 4x16 B tile (queries) with v_wmma_f32_16x16x4_f32, and fold
// the 16x16 f32 score tile into 8 per-lane running-min slots.
//
// f32 16x16x4 A operand layout (ISA 7.12.2): lane L holds A[M=L&15][K=2*(L>>4)]
// in VGPR0 and K+1 in VGPR1 -> one contiguous float2 from Apack.
// B (4x16) mirrors it: lane L holds B[K=2*(L>>4)..+1][N=L&15] -> contiguous
// float2 from Bpack (Bpack stored per-query row-major).
// C/D layout: VGPR c holds row M = c + 8*(L>=16), column N = L&15.
// ---------------------------------------------------------------------------
__global__ void __launch_bounds__(256)
nn_argmin_kernel(const float* __restrict__ Apack,
                 const float* __restrict__ Bpack,
                 int* __restrict__ inds) {
    const int lane = threadIdx.x & 31;
    const int wave = (blockIdx.x * blockDim.x + threadIdx.x) >> 5;
    const int qbase = wave * 16;
    const int col = lane & 15;   // query column within tile / A row within tile
    const int hi  = lane >> 4;   // K-half selector for A/B operands

    // Loop-invariant B operand (this wave's 16 query columns).
    v2f b = *(const v2f*)(Bpack + (qbase + col) * 4 + hi * 2);

    float bestv[8];
    int   bestr[8];
#pragma unroll
    for (int c = 0; c < 8; ++c) { bestv[c] = 3.4e38f; bestr[c] = 0; }

#pragma unroll 2
    for (int r = 0; r < NPTS; r += 16) {
        v2f a = *(const v2f*)(Apack + (r + col) * 4 + hi * 2);
        v8f cz = {};
        // D = A x B  (scores for refs r..r+15 vs this wave's 16 queries)
        v8f d = __builtin_amdgcn_wmma_f32_16x16x4_f32(
            /*neg_a=*/false, a, /*neg_b=*/false, b,
            /*c_mod=*/(short)0, cz, /*reuse_a=*/false, /*reuse_b=*/false);
#pragma unroll
        for (int c = 0; c < 8; ++c) {
            float s = d[c];
            // strict < keeps the earliest r (matches jnp.argmin tie rule
            // within this slot's residue class)
            if (s < bestv[c]) { bestv[c] = s; bestr[c] = r; }
        }
    }

    // Resolve slot -> absolute ref index, reduce 8 slots (tie: lower index).
    float bv = bestv[0];
    int   bi = bestr[0] + 8 * hi;  // slot 0 -> M = 0 + 8*hi
#pragma unroll
    for (int c = 1; c < 8; ++c) {
        int idx = bestr[c] + c + 8 * hi;
        bool take = (bestv[c] < bv) || ((bestv[c] == bv) && (idx < bi));
        bv = take ? bestv[c] : bv;
        bi = take ? idx : bi;
    }

    // Combine the lane pair (L covers rows 0-7, L+16 covers rows 8-15 of the
    // same query column) across the wave32.
    float ov = __shfl_xor(bv, 16, 32);
    int   oi = __shfl_xor(bi, 16, 32);
    bool take = (ov < bv) || ((ov == bv) && (oi < bi));
    bv = take ? ov : bv;
    bi = take ? oi : bi;

    if (lane < 16) inds[qbase + col] = bi;
}

// ---------------------------------------------------------------------------
// Kernel 3: assemble output [512, 8192]:
//   rows   0..255 : emb1 gathered at NN indices (emb1 is L2-resident, 8 MB)
//   rows 256..511 : emb2 streamed through
// ---------------------------------------------------------------------------
__global__ void nn_assemble_kernel(const float* __restrict__ emb1,
                                   const float* __restrict__ emb2,
                                   const int* __restrict__ inds,
                                   float* __restrict__ out) {
    int i = blockIdx.x * blockDim.x + threadIdx.x;  // 512*8192 elements
    int f = i >> 13;           // feature row (0..511)
    int n = i & (NPTS - 1);    // query column
    float v;
    if (f < NFEAT) {
        v = emb1[(f << 13) + inds[n]];
    } else {
        v = emb2[((f - NFEAT) << 13) + n];
    }
    out[i] = v;
}

extern "C" void kernel_launch(void* const* d_in, const int* in_sizes, int n_in,
                              void* d_out, int out_size, void* d_ws, size_t ws_size,
                              hipStream_t stream) {
    const float* emb1 = (const float*)d_in[0];  // [1,256,8192]
    const float* emb2 = (const float*)d_in[1];  // [1,256,8192]
    const float* t1   = (const float*)d_in[2];  // [1,3,8192]
    const float* t2   = (const float*)d_in[3];  // [1,3,8192]
    float* out = (float*)d_out;                 // [1,512,8192]

    // Workspace: A pack (128 KB) | B pack (128 KB) | indices (32 KB)
    float* Apack = (float*)d_ws;
    float* Bpack = Apack + NPTS * 4;
    int*   inds  = (int*)(Bpack + NPTS * 4);

    nn_prep_kernel<<<NPTS / 256, 256, 0, stream>>>(t1, t2, Apack, Bpack);

    // 8192 queries / 16 per wave = 512 waves; 8 waves (256 thr) per block.
    nn_argmin_kernel<<<64, 256, 0, stream>>>(Apack, Bpack, inds);

    // 512*8192 = 4194304 output elements.
    nn_assemble_kernel<<<(512 * NPTS) / 256, 256, 0, stream>>>(emb1, emb2, inds, out);
}